// Int8Linear_19662360281096
// MI455X (gfx1250) — compile-verified
//
#include <hip/hip_runtime.h>

// ---------------------------------------------------------------------------
// Int8Linear: out[16,11008] = x[16,4096] @ (W_int8 * scale).T + bias
//
// Memory-bound: 180 MB of int32 weights @ 23.3 TB/s => ~7.7 us floor.
// Exact-math path: acc = x @ float(W_int).T via V_WMMA_F32_16X16X4_F32
// (weights 0..126 are exact in f32), then epilogue applies per-row scale+bias.
// K split 8 ways -> 5504 waves for latency hiding; float atomics combine.
// ---------------------------------------------------------------------------

#define TOKENS 16
#define IN_F   4096
#define OUT_F  11008
#define KCHUNKS 8
#define KPER   (IN_F / KCHUNKS)   // 512 -> 128 WMMA steps per wave
#define WAVES_PER_BLOCK 8         // 256 threads (8 wave32)

typedef float v2f __attribute__((ext_vector_type(2)));
typedef float v8f __attribute__((ext_vector_type(8)));
typedef int   v2i __attribute__((ext_vector_type(2)));

__global__ void zero_out_kernel(float* __restrict__ out, int n) {
    int i = blockIdx.x * blockDim.x + threadIdx.x;
    if (i < n) out[i] = 0.0f;
}

__global__ __launch_bounds__(256) void int8linear_wmma_f32(
    const float* __restrict__ x,      // [16, 4096]
    const int*   __restrict__ w,      // [11008, 4096] int32 holding 0..126
    const float* __restrict__ scale,  // [11008]
    const float* __restrict__ bias,   // [11008]
    float*       __restrict__ out)    // [16, 11008]
{
    const int lane  = threadIdx.x & 31;
    const int wave  = threadIdx.x >> 5;
    const int ntile = blockIdx.x * WAVES_PER_BLOCK + wave;  // 0..687
    const int kbase = blockIdx.y * KPER;

    const int m     = lane & 15;       // token row (A) / column-in-tile (B)
    const int khalf = lane >> 4;       // 0: k={0,1}, 1: k={2,3} per WMMA step
    const int ncol  = ntile * 16 + m;  // output feature == weight row

    // Per-lane streaming pointers. Lane supplies k = kbase + 2*khalf + {0,1}
    // for both the A (x) and B (W^T) fragments -> one B64 load each per step.
    const float* xp = x + (size_t)m * IN_F + kbase + khalf * 2;
    const int*   wp = w + (size_t)ncol * IN_F + kbase + khalf * 2;

    v8f acc = {};   // 16x16 f32 C/D fragment (8 VGPRs)

    #pragma unroll 8
    for (int k = 0; k < KPER; k += 4) {
        v2f a = *(const v2f*)(xp + k);           // x[m][k0..k0+1]
        v2i bi = *(const v2i*)(wp + k);          // W[ncol][k0..k0+1]
        v2f b;
        b.x = (float)bi.x;                       // exact: values 0..126
        b.y = (float)bi.y;
        // D = A(16x4 f32) * B(4x16 f32) + C(16x16 f32)
        acc = __builtin_amdgcn_wmma_f32_16x16x4_f32(
            /*neg_a=*/false, a, /*neg_b=*/false, b,
            /*c_mod=*/(short)0, acc, /*reuse_a=*/false, /*reuse_b=*/false);
    }

    // Epilogue: C/D layout -> lane (L&15) = N col, VGPR r = row (L>=16 ? 8+r : r)
    const float s  = scale[ncol];
    const float bv = (blockIdx.y == 0) ? bias[ncol] : 0.0f;
    const int mbase = khalf * 8;
    #pragma unroll
    for (int r = 0; r < 8; ++r) {
        atomicAdd(&out[(size_t)(mbase + r) * OUT_F + ncol], acc[r] * s + bv);
    }
}

extern "C" void kernel_launch(void* const* d_in, const int* in_sizes, int n_in,
                              void* d_out, int out_size, void* d_ws, size_t ws_size,
                              hipStream_t stream) {
    const float* x     = (const float*)d_in[0];
    const int*   w     = (const int*)  d_in[1];
    const float* scale = (const float*)d_in[2];
    const float* bias  = (const float*)d_in[3];
    float*       out   = (float*)d_out;

    // d_out is poisoned; zero it before atomic accumulation (stream-ordered).
    zero_out_kernel<<<(out_size + 255) / 256, 256, 0, stream>>>(out, out_size);

    dim3 grid(OUT_F / (16 * WAVES_PER_BLOCK), KCHUNKS);   // (86, 8) -> 688 blocks
    int8linear_wmma_f32<<<grid, 256, 0, stream>>>(x, w, scale, bias, out);
}